// CombinedLoss_76063870812591
// MI455X (gfx1250) — compile-verified
//
#include <hip/hip_runtime.h>

typedef float v2f __attribute__((ext_vector_type(2)));
typedef float v8f __attribute__((ext_vector_type(8)));

#define BATCH   64
#define PPB     3072          // 3*32*32 preds per batch
#define KSEL    100
#define NTOT    6400          // BATCH*KSEL
#define STRIDE_F 32.0f        // 512/16
#define NEG_CNT 196608.0f     // 64*3*32*32

// workspace float offsets
#define WS_NEG   0
#define WS_SL1   1
#define WS_GB    2
#define OFF_PREDC 16
#define OFF_TGTC  (OFF_PREDC + NTOT*4)   // 25616
#define OFF_CONF  (OFF_TGTC + NTOT*4)    // 51216
#define OFF_MIOU  (OFF_CONF + NTOT)      // 57616
#define OFF_XP    (OFF_MIOU + NTOT)      // 64016: float2 per j = (x^2, (3.5-x)^2)

__device__ __forceinline__ float sigm(float v) { return 1.0f / (1.0f + expf(-v)); }

__device__ __forceinline__ unsigned lds_off(const void* p) {
    return (unsigned)(unsigned long long)p;   // low 32 bits of shared-aperture addr = LDS offset
}

__device__ __forceinline__ void async_ld_b32(unsigned lds, const float* gp) {
    asm volatile("global_load_async_to_lds_b32 %0, %1, off"
                 :: "v"(lds), "v"(gp) : "memory");
}
__device__ __forceinline__ void async_ld_b64(unsigned lds, const float* gp) {
    asm volatile("global_load_async_to_lds_b64 %0, %1, off"
                 :: "v"(lds), "v"(gp) : "memory");
}
__device__ __forceinline__ void async_ld_b128(unsigned lds, const float* gp) {
    asm volatile("global_load_async_to_lds_b128 %0, %1, off"
                 :: "v"(lds), "v"(gp) : "memory");
}
__device__ __forceinline__ void wait_async0() {
    asm volatile("s_wait_asynccnt 0x0" ::: "memory");
}

// Full-f32 32-lane sum via V_WMMA_F32_16X16X4_F32:
// A(16x4) holds lane partials in K=0 (lanes 0-15) / K=2 (lanes 16-31), B = ones.
// Every lane's 8 D values sum to rows 0-7 (lanes 0-15) or rows 8-15 (lanes 16-31);
// lane 0 + lane 16 together hold the full wave sum.
// Caller must guarantee EXEC is all 1s (no divergence at call site).
__device__ __forceinline__ float wave_sum_wmma(float acc) {
    v2f A; A.x = acc; A.y = 0.0f;
    v2f Bm; Bm.x = 1.0f; Bm.y = 1.0f;
    v8f C = {};
    v8f D = __builtin_amdgcn_wmma_f32_16x16x4_f32(false, A, false, Bm,
                                                  (short)0, C, false, false);
    return D[0]+D[1]+D[2]+D[3]+D[4]+D[5]+D[6]+D[7];
}

__device__ __forceinline__ void wave_atomic_sum(float acc, float* dst) {
    float w = wave_sum_wmma(acc);
    int lane = threadIdx.x & 31;
    if (lane == 0 || lane == 16) atomicAdd(dst, w);
}

__global__ void k_init(float* ws) {
    if (threadIdx.x < 3) ws[threadIdx.x] = 0.0f;
}

// ---- negative-size penalty on RAW pred w/h --------------------------------
// grid 192x256 -> exactly 4 iterations/thread, no tail divergence (WMMA-safe)
__global__ void k_negpen(const float* __restrict__ pred, float* ws) {
    float acc = 0.0f;
    for (int e = blockIdx.x * 256 + threadIdx.x; e < 196608; e += gridDim.x * 256) {
        __builtin_prefetch(pred + e * 5 + 2560, 0, 0);
        float pw = pred[e * 5 + 2], ph = pred[e * 5 + 3];
        acc += fmaxf(1.0f - pw, 0.0f) + fmaxf(1.0f - ph, 0.0f);
    }
    wave_atomic_sum(acc, ws + WS_NEG);
}

// ---- fix target coords ----------------------------------------------------
__global__ void k_tgtfix(const float* __restrict__ tb, float* ws) {
    int j = blockIdx.x * 256 + threadIdx.x;
    if (j >= NTOT) return;
    float t0 = tb[j*4+0], t1 = tb[j*4+1], t2 = tb[j*4+2], t3 = tb[j*4+3];
    float x1 = fminf(t0, t2), y1 = fminf(t1, t3);
    float x2 = fmaxf(t0, t2), y2 = fmaxf(t1, t3);
    if (x1 == x2) x2 = x1 + 1.0f;
    if (y1 == y2) y2 = y1 + 1.0f;
    float* o = ws + OFF_TGTC + j*4;
    o[0] = x1; o[1] = y1; o[2] = x2; o[3] = y2;
}

// ---- per-batch top-100 by conf + postprocess selected ---------------------
__global__ void k_topk(const float* __restrict__ pred, float* ws) {
    __shared__ float vals[PPB];
    __shared__ float rv[256];
    __shared__ int   ri[256];
    __shared__ int   widx[KSEL];
    __shared__ float wval[KSEL];
    const int b = blockIdx.x, tid = threadIdx.x;
    const float* pb = pred + (size_t)b * PPB * 5;

    for (int i = tid; i < PPB; i += 256) vals[i] = sigm(pb[i * 5 + 4]);
    __syncthreads();

    for (int r = 0; r < KSEL; r++) {
        float bv = -1e30f; int bi = 0;
        for (int i = tid; i < PPB; i += 256) {
            float v = vals[i];
            if (v > bv) { bv = v; bi = i; }
        }
        rv[tid] = bv; ri[tid] = bi; __syncthreads();
        for (int s = 128; s > 0; s >>= 1) {
            if (tid < s) {
                float v2 = rv[tid + s]; int i2 = ri[tid + s];
                if (v2 > rv[tid] || (v2 == rv[tid] && i2 < ri[tid])) { rv[tid] = v2; ri[tid] = i2; }
            }
            __syncthreads();
        }
        if (tid == 0) { widx[r] = ri[0]; wval[r] = rv[0]; vals[ri[0]] = -1e30f; }
        __syncthreads();
    }

    if (tid < KSEL) {
        int idx = widx[tid];
        int rem = idx & 1023, hh = rem >> 5, wc = rem & 31;
        const float* p = pb + (size_t)idx * 5;
        float x = (sigm(p[0]) + (float)wc) * STRIDE_F;
        float y = (sigm(p[1]) + (float)hh) * STRIDE_F;
        float w = expf(p[2]) * STRIDE_F;
        float h = expf(p[3]) * STRIDE_F;
        int row = b * KSEL + tid;
        float* o = ws + OFF_PREDC + row * 4;
        o[0] = x - 0.5f * w; o[1] = y - 0.5f * h;
        o[2] = x + 0.5f * w; o[3] = y + 0.5f * h;
        ws[OFF_CONF + row] = wval[tid];
    }
}

// ---- smooth-L1 sum over pred_c vs target_c --------------------------------
// grid 25x256 = 6400 threads, 25600 elems -> exactly 4 iters/thread (WMMA-safe)
__global__ void k_sl1(float* ws) {
    float acc = 0.0f;
    for (int e = blockIdx.x * 256 + threadIdx.x; e < NTOT * 4; e += gridDim.x * 256) {
        float d = ws[OFF_PREDC + e] - ws[OFF_TGTC + e];
        float ad = fabsf(d);
        acc += (ad < 1.0f) ? 0.5f * d * d : (ad - 0.5f);
    }
    wave_atomic_sum(acc, ws + WS_SL1);
}

// ---- per-target column max of 6400x6400 IoU matrix ------------------------
// block = 64 targets x all preds; targets + pred chunks staged via CDNA5
// async global->LDS (b32 tile / b128 rows), inner loop is LDS-only.
__global__ void k_ioumax(float* ws) {
    __shared__ float sT[256];     // 64 targets * 4
    __shared__ float sP[1024];    // 256 preds * 4 per chunk
    __shared__ float sred[256];
    const int tid = threadIdx.x;
    const int j0 = blockIdx.x * 64;

    async_ld_b32(lds_off(&sT[tid]), ws + OFF_TGTC + j0 * 4 + tid);
    wait_async0();
    __syncthreads();

    const int jl = tid & 63, g = tid >> 6;
    const float tx1 = sT[jl*4], ty1 = sT[jl*4+1], tx2 = sT[jl*4+2], ty2 = sT[jl*4+3];
    const float ta = (tx2 - tx1) * (ty2 - ty1);
    float best = -1e30f;

    for (int c = 0; c < NTOT / 256; c++) {          // 25 chunks of 256 preds
        __syncthreads();                            // prior chunk fully consumed
        async_ld_b128(lds_off(&sP[tid * 4]), ws + OFF_PREDC + (c * 256 + tid) * 4);
        wait_async0();
        __syncthreads();
        const float* base = &sP[g * 256];           // this group's 64 rows
        #pragma unroll 4
        for (int k = 0; k < 64; k++) {
            float px1 = base[k*4+0], py1 = base[k*4+1];
            float px2 = base[k*4+2], py2 = base[k*4+3];
            float pa = (px2 - px1) * (py2 - py1);
            float iw = fmaxf(fminf(px2, tx2) - fmaxf(px1, tx1), 0.0f);
            float ih = fmaxf(fminf(py2, ty2) - fmaxf(py1, ty1), 0.0f);
            float inter = iw * ih;
            best = fmaxf(best, inter / (pa + ta - inter));
        }
    }
    sred[tid] = best; __syncthreads();
    if (g == 0) {
        float m = fmaxf(fmaxf(sred[jl], sred[jl + 64]),
                        fmaxf(sred[jl + 128], sred[jl + 192]));
        ws[OFF_MIOU + j0 + jl] = m;
    }
}

// ---- x[j] = iou_loss*2 + diou + sl1; store (x^2, (3.5-x)^2) pair ----------
__global__ void k_x(float* ws) {
    int j = blockIdx.x * 256 + threadIdx.x;
    if (j >= NTOT) return;
    float sl1 = ws[WS_SL1] / (NTOT * 4.0f) / 512.0f;
    float mi = ws[OFF_MIOU + j];
    if (!(mi == mi)) mi = 0.0f;          // nan_to_num
    mi = fmaxf(mi, 0.0f);                // clip
    const float* p = ws + OFF_PREDC + j * 4;
    const float* t = ws + OFF_TGTC + j * 4;
    float pa = fmaxf(p[2]-p[0],0.f) * fmaxf(p[3]-p[1],0.f);
    float ta = fmaxf(t[2]-t[0],0.f) * fmaxf(t[3]-t[1],0.f);
    float inter = fmaxf(fminf(p[2],t[2]) - fmaxf(p[0],t[0]), 0.f) *
                  fmaxf(fminf(p[3],t[3]) - fmaxf(p[1],t[1]), 0.f);
    float iou = inter / (pa + ta - inter + 1e-7f);
    float dx = 0.5f*(p[0]+p[2]) - 0.5f*(t[0]+t[2]);
    float dy = 0.5f*(p[1]+p[3]) - 0.5f*(t[1]+t[3]);
    float cdist = dx*dx + dy*dy;
    float ew = fmaxf(p[2],t[2]) - fminf(p[0],t[0]);
    float eh = fmaxf(p[3],t[3]) - fminf(p[1],t[1]);
    float diag = ew*ew + eh*eh;
    float diou = 1.0f - (iou - cdist / (diag + 1e-7f));
    float x = (1.0f - mi) * 2.0f + diou + sl1;
    float xm = 3.5f - x;
    float* o = ws + OFF_XP + j * 2;
    o[0] = x * x;
    o[1] = xm * xm;
}

// ---- good-bad double sum over 6400 x 6400 ---------------------------------
// grid (25, 5): block.x covers 256 i-rows (conf), block.y a 1280-wide j chunk.
// Precomputed (x^2,(3.5-x)^2) pairs staged via async b64 -> LDS; per-element
// work = ds_load_b64 + 2 add + 2 v_sqrt_f32 + 2 fma (sqrt-throughput bound).
// Wave partials reduced with full-f32 V_WMMA_F32_16X16X4_F32.
#define JCHUNK 1280
__global__ void k_goodbad(float* ws) {
    __shared__ float2 sxp[JCHUNK];
    const int tid = threadIdx.x;
    const int i = blockIdx.x * 256 + tid;         // < 6400 by construction
    const int j0 = blockIdx.y * JCHUNK;

    const float* xp = ws + OFF_XP + j0 * 2;
    for (int k = tid; k < JCHUNK; k += 256)       // 5 b64 loads per thread
        async_ld_b64(lds_off(&sxp[k]), xp + k * 2);
    wait_async0();
    __syncthreads();

    float y = ws[OFF_CONF + i];
    float byy = (1.0f - y) * (1.0f - y);
    float acc = 0.0f;
    #pragma unroll 8
    for (int j = 0; j < JCHUNK; j++) {
        float2 a = sxp[j];
        acc += 2.0f * sqrtf(a.x + byy) - 1.5f * sqrtf(a.y + byy);
    }
    wave_atomic_sum(acc, ws + WS_GB);
}

// ---- final combine --------------------------------------------------------
__global__ void k_final(const float* __restrict__ ws, float* out) {
    float neg = ws[WS_NEG] / NEG_CNT;
    float gb = ws[WS_GB] / ((float)NTOT * (float)NTOT);
    float losses = fmaxf(gb + 3.5f * 1.5f, 0.0f);   // + max_loss*BAD_MULT
    out[0] = losses + neg;
}

extern "C" void kernel_launch(void* const* d_in, const int* in_sizes, int n_in,
                              void* d_out, int out_size, void* d_ws, size_t ws_size,
                              hipStream_t stream) {
    const float* pred = (const float*)d_in[0];   // [64,3,32,32,5]
    const float* tgt  = (const float*)d_in[1];   // [64,100,4]
    float* ws  = (float*)d_ws;
    float* out = (float*)d_out;

    k_init   <<<1, 64, 0, stream>>>(ws);
    k_negpen <<<192, 256, 0, stream>>>(pred, ws);
    k_tgtfix <<<(NTOT + 255) / 256, 256, 0, stream>>>(tgt, ws);
    k_topk   <<<BATCH, 256, 0, stream>>>(pred, ws);
    k_sl1    <<<25, 256, 0, stream>>>(ws);
    k_ioumax <<<NTOT / 64, 256, 0, stream>>>(ws);
    k_x      <<<(NTOT + 255) / 256, 256, 0, stream>>>(ws);
    k_goodbad<<<dim3(NTOT / 256, NTOT / JCHUNK), 256, 0, stream>>>(ws);
    k_final  <<<1, 1, 0, stream>>>(ws, out);
}